// MedianBlur_55233279427167
// MI455X (gfx1250) — compile-verified
//
#include <hip/hip_runtime.h>

// Native clang vector type: valid for __builtin_nontemporal_store/load and
// lowers to b128 global memory ops.
typedef float v4f __attribute__((ext_vector_type(4)));

// -----------------------------------------------------------------------------
// Exact median-of-9 via min/max exchange network (Smith / McGuire, 20 exchanges).
// Produces sorted[4] of the 9 inputs == jnp.median for 9 elements.
// -----------------------------------------------------------------------------
#define S2(a, b)                \
    do {                        \
        float _t = fminf(a, b); \
        b = fmaxf(a, b);        \
        a = _t;                 \
    } while (0)

#define MN3(a, b, c) do { S2(a, b); S2(a, c); } while (0)
#define MX3(a, b, c) do { S2(b, c); S2(a, c); } while (0)
#define MNMX3(a, b, c)          do { MX3(a, b, c); S2(a, b); } while (0)
#define MNMX4(a, b, c, d)       do { S2(a, b); S2(c, d); S2(a, c); S2(b, d); } while (0)
#define MNMX5(a, b, c, d, e)    do { S2(a, b); S2(c, d); MN3(a, c, e); MX3(b, d, e); } while (0)
#define MNMX6(a, b, c, d, e, f) do { S2(a, d); S2(b, e); S2(c, f); MN3(a, b, c); MX3(d, e, f); } while (0)

__device__ __forceinline__ float median9(float v0, float v1, float v2,
                                         float v3, float v4, float v5,
                                         float v6, float v7, float v8) {
    MNMX6(v0, v1, v2, v3, v4, v5);
    MNMX5(v1, v2, v3, v4, v6);
    MNMX4(v2, v3, v4, v7);
    MNMX3(v3, v4, v8);
    return v4;
}

// -----------------------------------------------------------------------------
// 3x3 median blur, zero padding, f32, W multiple of 4.
// Each thread: one output row, 4 consecutive pixels (one b128 store).
// Per contributing row: 1x global_load_b128 + 2 scalar halo loads.
// Block (128, 2): full 512-wide row pair per block (vertical cache locality).
// -----------------------------------------------------------------------------
__global__ __launch_bounds__(256) void median_blur3x3_kernel(
    const float* __restrict__ in, float* __restrict__ out, int H, int W) {
    const int qx = threadIdx.x;                      // quad index within row
    const int x0 = qx << 2;                          // first pixel of quad
    const int y  = blockIdx.x * blockDim.y + threadIdx.y;
    const long long img    = blockIdx.y;             // flattened (b, c)
    const long long planeo = img * (long long)H * (long long)W;

    const float* base = in + planeo;

    // r[k][0..5] = columns x0-1 .. x0+4 of row y+k-1 (zero-padded)
    float r[3][6];
#pragma unroll
    for (int k = 0; k < 3; ++k) {
        const int yy = y + k - 1;
        if (yy < 0 || yy >= H) {
#pragma unroll
            for (int j = 0; j < 6; ++j) r[k][j] = 0.0f;
        } else {
            const float* rp = base + (long long)yy * W;
            r[k][0] = (x0 > 0) ? rp[x0 - 1] : 0.0f;                   // left halo
            const v4f c = *reinterpret_cast<const v4f*>(rp + x0);     // b128 load
            r[k][1] = c.x; r[k][2] = c.y; r[k][3] = c.z; r[k][4] = c.w;
            r[k][5] = (x0 + 4 < W) ? rp[x0 + 4] : 0.0f;               // right halo
        }
    }

    float res[4];
#pragma unroll
    for (int i = 0; i < 4; ++i) {
        res[i] = median9(r[0][i], r[0][i + 1], r[0][i + 2],
                         r[1][i], r[1][i + 1], r[1][i + 2],
                         r[2][i], r[2][i + 1], r[2][i + 2]);
    }

    v4f o;
    o.x = res[0]; o.y = res[1]; o.z = res[2]; o.w = res[3];
    // Output is never re-read: non-temporal b128 store keeps input rows in cache.
    v4f* dst = reinterpret_cast<v4f*>(out + planeo + (long long)y * W + x0);
    __builtin_nontemporal_store(o, dst);
}

// -----------------------------------------------------------------------------
// Harness entry point.
// in_sizes[0] = 16*3*512*512 f32; output same shape.
// -----------------------------------------------------------------------------
extern "C" void kernel_launch(void* const* d_in, const int* in_sizes, int n_in,
                              void* d_out, int out_size, void* d_ws, size_t ws_size,
                              hipStream_t stream) {
    (void)n_in; (void)out_size; (void)d_ws; (void)ws_size;
    const float* x   = (const float*)d_in[0];
    float*       out = (float*)d_out;

    const int H = 512;
    const int W = 512;
    const int images = in_sizes[0] / (H * W);   // 16 * 3 = 48 planes

    dim3 block(W / 4, 2, 1);                    // (128, 2) = 256 threads, 8 waves
    dim3 grid(H / 2, images, 1);                // (256, 48)
    median_blur3x3_kernel<<<grid, block, 0, stream>>>(x, out, H, W);
}